// MultiRelGNN_54812372631715
// MI455X (gfx1250) — compile-verified
//
#include <hip/hip_runtime.h>
#include <math.h>

typedef __attribute__((ext_vector_type(2))) float v2f;
typedef __attribute__((ext_vector_type(8))) float v8f;

#define TFIN 128
#define THID 128
#define LN_EPS 1e-5f

#if defined(__HIP_DEVICE_COMPILE__) && defined(__gfx1250__) && \
    __has_builtin(__builtin_amdgcn_global_load_async_to_lds_b128) && \
    __has_builtin(__builtin_amdgcn_s_wait_asynccnt)
#define USE_ASYNC_LDS 1
#define GLOBAL_AS __attribute__((address_space(1)))
#define LDS_AS __attribute__((address_space(3)))
typedef __attribute__((__vector_size__(4 * sizeof(int)))) int b128_t;
#endif

// ---------------------------------------------------------------------------
// degree / normalization helpers
// ---------------------------------------------------------------------------
__global__ void k_fill1(float* __restrict__ p, int n) {
  int i = blockIdx.x * blockDim.x + threadIdx.x;
  if (i < n) p[i] = 1.0f;  // self-loop contributes 1 to every degree
}

__global__ void k_deg(const int* __restrict__ dst, int e, float* __restrict__ deg) {
  int i = blockIdx.x * blockDim.x + threadIdx.x;
  if (i < e) atomicAdd(&deg[dst[i]], 1.0f);
}

__global__ void k_rsqrt_inplace(float* __restrict__ p, int n) {
  int i = blockIdx.x * blockDim.x + threadIdx.x;
  if (i < n) p[i] = rsqrtf(p[i]);  // deg >= 1 always (self-loops)
}

// ---------------------------------------------------------------------------
// WMMA fp32 GEMM:  out[N, NC] = concat(X0, X1)[N, K] @ W[K, NC]
// One wave32 computes one 16x16 output tile via V_WMMA_F32_16X16X4_F32.
// Tiles staged to LDS via GLOBAL_LOAD_ASYNC_TO_LDS_B128 when available.
// OOB rows of the X tile are left unwritten: row m of A only affects row m
// of D, which is masked on the store side, and W is always fully valid.
// ---------------------------------------------------------------------------
template <int K, int K0, int NC>
__launch_bounds__(32 * (NC / 16))
__global__ void k_gemm_wmma(const float* __restrict__ X0, const float* __restrict__ X1,
                            const float* __restrict__ W, float* __restrict__ out, int n) {
  extern __shared__ float lds[];
  float* sX = lds;            // 16 * K
  float* sW = lds + 16 * K;   // K * NC
  const int tid = threadIdx.x;
  const int m0 = blockIdx.x * 16;

  // ---- stage X tile (16 x K) ----
  constexpr int XG = 16 * (K / 4);  // float4 granules
  for (int idx = tid; idx < XG; idx += blockDim.x) {
    const int r = idx / (K / 4);
    const int k4 = (idx - r * (K / 4)) * 4;
    if (m0 + r < n) {
      const float* gp = (k4 < K0) ? &X0[(long)(m0 + r) * K0 + k4]
                                  : &X1[(long)(m0 + r) * K0 + (k4 - K0)];
      float* lp = &sX[r * K + k4];
#ifdef USE_ASYNC_LDS
      __builtin_amdgcn_global_load_async_to_lds_b128(
          (GLOBAL_AS b128_t*)gp, (LDS_AS b128_t*)lp, /*offset=*/0, /*cpol=*/0);
#else
      *(float4*)lp = *(const float4*)gp;
#endif
    }
  }
  // ---- stage W (K x NC) ----
  constexpr int WG = (K * NC) / 4;
  for (int idx = tid; idx < WG; idx += blockDim.x) {
    const float* gp = &W[idx * 4];
    float* lp = &sW[idx * 4];
#ifdef USE_ASYNC_LDS
    __builtin_amdgcn_global_load_async_to_lds_b128(
        (GLOBAL_AS b128_t*)gp, (LDS_AS b128_t*)lp, 0, 0);
#else
    *(float4*)lp = *(const float4*)gp;
#endif
  }
#ifdef USE_ASYNC_LDS
  __builtin_amdgcn_s_wait_asynccnt(0);
#endif
  __syncthreads();

  const int lane = tid & 31;
  const int wave = tid >> 5;
  const int row  = lane & 15;          // A-matrix M row held by this lane
  const int col  = lane & 15;          // B/C/D N column held by this lane
  const int koff = (lane >> 4) * 2;    // lanes 16-31 hold K=2,3 of each step
  const int cn   = wave * 16;

  v8f acc = {};
#pragma unroll 4
  for (int kk = 0; kk < K; kk += 4) {
    const int k0 = kk + koff;
    v2f a, b;
    a.x = sX[row * K + k0];
    a.y = sX[row * K + k0 + 1];
    b.x = sW[k0 * NC + cn + col];
    b.y = sW[(k0 + 1) * NC + cn + col];
    acc = __builtin_amdgcn_wmma_f32_16x16x4_f32(
        /*neg_a=*/false, a, /*neg_b=*/false, b,
        /*c_mod=*/(short)0, acc, /*reuse_a=*/false, /*reuse_b=*/false);
  }

  if (m0 + 16 <= n) {
    // fast path: whole tile in range, no per-element exec juggling
#pragma unroll
    for (int r = 0; r < 8; ++r) {
      const int orow = r + (lane >> 4) * 8;  // C/D layout: VGPR r -> M=r / M=r+8
      out[(long)(m0 + orow) * NC + cn + col] = acc[r];
    }
  } else {
#pragma unroll
    for (int r = 0; r < 8; ++r) {
      const int orow = r + (lane >> 4) * 8;
      if (m0 + orow < n) out[(long)(m0 + orow) * NC + cn + col] = acc[r];
    }
  }
}

// ---------------------------------------------------------------------------
// agg[i,c] = h[i,c] * dis[i]^2 + bias[c]   (self-loop term + bias as init)
// ---------------------------------------------------------------------------
__global__ void k_selfbias(const float* __restrict__ h, const float* __restrict__ dis,
                           const float* __restrict__ bias, float* __restrict__ out,
                           int n, int C) {
  long i = (long)blockIdx.x * blockDim.x + threadIdx.x;
  if (i >= (long)n * C) return;
  int r = (int)(i / C);
  int c = (int)(i - (long)r * C);
  float d = dis[r];
  out[i] = h[i] * d * d + bias[c];
}

// ---------------------------------------------------------------------------
// Scatter-add edges: out[dst] += h[src] * dis[src]*dis[dst]
// One thread = (edge, 4-feature group); atomics resolve in L2 (data resident).
// ---------------------------------------------------------------------------
__global__ void k_scatter(const int* __restrict__ src, const int* __restrict__ dst, int e,
                          const float* __restrict__ h, const float* __restrict__ dis,
                          float* __restrict__ out, int C) {
  const int c4 = C >> 2;
  const long total = (long)e * c4;
  for (long i = (long)blockIdx.x * blockDim.x + threadIdx.x; i < total;
       i += (long)gridDim.x * blockDim.x) {
    const int eidx = (int)(i / c4);
    const int c = (int)(i - (long)eidx * c4) * 4;
    const int s = src[eidx];
    const int d = dst[eidx];
    const float norm = dis[s] * dis[d];
    const float4 hv = *(const float4*)(h + (long)s * C + c);
    float* o = out + (long)d * C + c;
    atomicAdd(o + 0, hv.x * norm);
    atomicAdd(o + 1, hv.y * norm);
    atomicAdd(o + 2, hv.z * norm);
    atomicAdd(o + 3, hv.w * norm);
  }
}

// ---------------------------------------------------------------------------
// LayerNorm + ELU, one wave32 per row of width C; writes into out with
// row stride outStride at column offset outColOff (fuses the concat).
// ---------------------------------------------------------------------------
template <int C>
__global__ void k_ln_elu(const float* __restrict__ in, const float* __restrict__ g,
                         const float* __restrict__ b, float* __restrict__ out,
                         int n, int outStride, int outColOff) {
  const int lane = threadIdx.x & 31;
  const int row = (blockIdx.x * blockDim.x + threadIdx.x) >> 5;
  if (row >= n) return;
  constexpr int PER = C / 32;
  float v[PER];
  float s = 0.0f;
#pragma unroll
  for (int j = 0; j < PER; ++j) {
    v[j] = in[(long)row * C + lane * PER + j];
    s += v[j];
  }
#pragma unroll
  for (int off = 16; off > 0; off >>= 1) s += __shfl_xor(s, off, 32);
  const float m = s * (1.0f / C);
  float vs = 0.0f;
#pragma unroll
  for (int j = 0; j < PER; ++j) {
    const float d = v[j] - m;
    vs += d * d;
  }
#pragma unroll
  for (int off = 16; off > 0; off >>= 1) vs += __shfl_xor(vs, off, 32);
  const float inv = rsqrtf(vs * (1.0f / C) + LN_EPS);
#pragma unroll
  for (int j = 0; j < PER; ++j) {
    const int c = lane * PER + j;
    float y = (v[j] - m) * inv * g[c] + b[c];
    y = (y > 0.0f) ? y : (expf(y) - 1.0f);   // ELU
    out[(long)row * outStride + outColOff + c] = y;
  }
}

// ---------------------------------------------------------------------------
// out[i] = h[i,:] @ W_out + b_out, one wave32 per row
// ---------------------------------------------------------------------------
__global__ void k_out(const float* __restrict__ h, const float* __restrict__ w,
                      const float* __restrict__ b, float* __restrict__ out, int n) {
  const int lane = threadIdx.x & 31;
  const int row = (blockIdx.x * blockDim.x + threadIdx.x) >> 5;
  if (row >= n) return;
  float s = 0.0f;
#pragma unroll
  for (int j = 0; j < 4; ++j) {
    const int c = lane * 4 + j;
    s += h[(long)row * THID + c] * w[c];
  }
#pragma unroll
  for (int off = 16; off > 0; off >>= 1) s += __shfl_xor(s, off, 32);
  if (lane == 0) out[row] = s + b[0];
}

// ---------------------------------------------------------------------------
extern "C" void kernel_launch(void* const* d_in, const int* in_sizes, int n_in,
                              void* d_out, int out_size, void* d_ws, size_t ws_size,
                              hipStream_t stream) {
  const float* x      = (const float*)d_in[0];
  const int*   eC     = (const int*)d_in[1];
  const int*   eV     = (const int*)d_in[2];
  const float* xlag   = (const float*)d_in[3];
  const float* W_corr = (const float*)d_in[4];
  const float* b_corr = (const float*)d_in[5];
  const float* g_lnC  = (const float*)d_in[6];
  const float* b_lnC  = (const float*)d_in[7];
  const float* W_vend = (const float*)d_in[8];
  const float* b_vend = (const float*)d_in[9];
  const float* g_lnV  = (const float*)d_in[10];
  const float* b_lnV  = (const float*)d_in[11];
  const float* W_ref  = (const float*)d_in[12];
  const float* b_ref  = (const float*)d_in[13];
  const float* g_lnR  = (const float*)d_in[14];
  const float* b_lnR  = (const float*)d_in[15];
  const float* W_out  = (const float*)d_in[16];
  const float* b_out  = (const float*)d_in[17];

  const int N = in_sizes[0] / TFIN;
  const int E = in_sizes[1] / 2;

  const int* srcC = eC;
  const int* dstC = eC + E;
  const int* srcV = eV;
  const int* dstV = eV + E;

  // workspace layout (floats)
  float* ws   = (float*)d_ws;
  float* disC = ws;                    // N
  float* disV = disC + N;              // N
  float* disA = disV + N;              // N
  float* bufT = disA + N;              // N*128 (GEMM outputs h1/h2/h3, then final LN)
  float* bufG = bufT + (long)N * THID; // N*128 (aggregation accumulator)
  float* bufH = bufG + (long)N * THID; // N*128 (concat hidden)

  auto cdiv = [](long a, long b) { return (int)((a + b - 1) / b); };
  const int mt = cdiv(N, 16);  // M tiles for GEMM

  // --- degrees: deg = 1 (self loop) + in-degree; dis = rsqrt(deg) ---
  k_fill1<<<cdiv(N, 256), 256, 0, stream>>>(disC, N);
  k_fill1<<<cdiv(N, 256), 256, 0, stream>>>(disV, N);
  k_fill1<<<cdiv(N, 256), 256, 0, stream>>>(disA, N);
  k_deg<<<cdiv(E, 256), 256, 0, stream>>>(dstC, E, disC);
  k_deg<<<cdiv(E, 256), 256, 0, stream>>>(dstV, E, disV);
  k_deg<<<cdiv(E, 256), 256, 0, stream>>>(dstC, E, disA);
  k_deg<<<cdiv(E, 256), 256, 0, stream>>>(dstV, E, disA);
  k_rsqrt_inplace<<<cdiv(N, 256), 256, 0, stream>>>(disC, N);
  k_rsqrt_inplace<<<cdiv(N, 256), 256, 0, stream>>>(disV, N);
  k_rsqrt_inplace<<<cdiv(N, 256), 256, 0, stream>>>(disA, N);

  // --- layer 1: corr branch (K=128 -> 64) ---
  k_gemm_wmma<128, 128, 64><<<mt, 128, (16 * 128 + 128 * 64) * 4, stream>>>(
      x, x, W_corr, bufT, N);
  k_selfbias<<<cdiv((long)N * 64, 256), 256, 0, stream>>>(bufT, disC, b_corr, bufG, N, 64);
  k_scatter<<<cdiv((long)E * 16, 256), 256, 0, stream>>>(srcC, dstC, E, bufT, disC, bufG, 64);
  k_ln_elu<64><<<cdiv(N, 8), 256, 0, stream>>>(bufG, g_lnC, b_lnC, bufH, N, THID, 0);

  // --- layer 1: vendor branch (K=256 concat -> 64) ---
  k_gemm_wmma<256, 128, 64><<<mt, 128, (16 * 256 + 256 * 64) * 4, stream>>>(
      x, xlag, W_vend, bufT, N);
  k_selfbias<<<cdiv((long)N * 64, 256), 256, 0, stream>>>(bufT, disV, b_vend, bufG, N, 64);
  k_scatter<<<cdiv((long)E * 16, 256), 256, 0, stream>>>(srcV, dstV, E, bufT, disV, bufG, 64);
  k_ln_elu<64><<<cdiv(N, 8), 256, 0, stream>>>(bufG, g_lnV, b_lnV, bufH, N, THID, 64);

  // --- layer 2: refine on combined graph (K=128 -> 128) ---
  k_gemm_wmma<128, 128, 128><<<mt, 256, (16 * 128 + 128 * 128) * 4, stream>>>(
      bufH, bufH, W_ref, bufT, N);
  k_selfbias<<<cdiv((long)N * 128, 256), 256, 0, stream>>>(bufT, disA, b_ref, bufG, N, 128);
  k_scatter<<<cdiv((long)E * 32, 256), 256, 0, stream>>>(srcC, dstC, E, bufT, disA, bufG, 128);
  k_scatter<<<cdiv((long)E * 32, 256), 256, 0, stream>>>(srcV, dstV, E, bufT, disA, bufG, 128);
  k_ln_elu<128><<<cdiv(N, 8), 256, 0, stream>>>(bufG, g_lnR, b_lnR, bufT, N, THID, 0);

  // --- output head ---
  k_out<<<cdiv(N, 8), 256, 0, stream>>>(bufT, W_out, b_out, (float*)d_out, N);
}